// DMAF_12197707120713
// MI455X (gfx1250) — compile-verified
//
#include <hip/hip_runtime.h>

// ---------------------------------------------------------------------------
// MI455X / gfx1250 implementation.
//
// Algebraic restructuring (exact, not approximate):
//  * cross_attn #3 has a single key -> softmax == 1 -> Wk3/Wq3 are dead;
//    out = x + broadcast( (ln3(att) @ Wv3) @ Wo3 + bo3 ).
//  * cross_attn #1 has a single query per (p,head) -> dots_j = ln(x_j)·U[p]
//    with U[p] = SCALE * Wk1 @ q[p]  (256x8), and
//    o_head = (sum_j a_j ln(x_j)) @ Wv1_head.
//  -> the two 262144x256x512 projections become memory-bound row passes.
//  HBM traffic ~= 4 reads + 1 write of 256MB ~= 1.25GB -> ~55us @ 23.3TB/s.
//  fp32 kept end-to-end; matrix work done with V_WMMA_F32_16X16X4_F32.
// ---------------------------------------------------------------------------

typedef float v2f __attribute__((ext_vector_type(2)));
typedef float v8f __attribute__((ext_vector_type(8)));

#define B_SZ   32
#define P_SZ   4
#define N_SZ   2048
#define D_SZ   256
#define INNER  512
#define HEADS  8
#define DH     64
#define SCALE_F 0.125f      // 64^-0.5
#define EPS_F   1e-3f

__device__ __forceinline__ float wave_sum(float v) {
#pragma unroll
  for (int m = 16; m >= 1; m >>= 1) v += __shfl_xor(v, m, 32);
  return v;
}
__device__ __forceinline__ float wave_max(float v) {
#pragma unroll
  for (int m = 16; m >= 1; m >>= 1) v = fmaxf(v, __shfl_xor(v, m, 32));
  return v;
}

// ---------------------------------------------------------------------------
// K0: gft = gfe@Wg + bg ; U[p] = SCALE * Wk1 @ (ln1(gft[p]) @ Wq1) per head.
// U stored padded to 16 columns (heads 8..15 = 0) for WMMA N=16.
// ---------------------------------------------------------------------------
__global__ __launch_bounds__(256) void k0_setup(
    const float* __restrict__ gfe, const float* __restrict__ Wg,
    const float* __restrict__ bg,  const float* __restrict__ g1,
    const float* __restrict__ b1,  const float* __restrict__ Wk1,
    const float* __restrict__ Wq1,
    float* __restrict__ U, float* __restrict__ gft_out)
{
  __shared__ float gft[P_SZ][D_SZ];
  __shared__ float lng[P_SZ][D_SZ];
  __shared__ float qh[P_SZ][INNER];
  __shared__ float st[P_SZ][2];
  const int tid = threadIdx.x;

  for (int o = tid; o < P_SZ * D_SZ; o += 256) {
    int p = o >> 8, d = o & 255;
    float s = bg[d];
    for (int c = 0; c < D_SZ; ++c) s += gfe[p * D_SZ + c] * Wg[c * D_SZ + d];
    gft[p][d] = s;
    gft_out[o] = s;
  }
  __syncthreads();
  if (tid < P_SZ) {
    float s = 0.f, s2 = 0.f;
    for (int d = 0; d < D_SZ; ++d) { float v = gft[tid][d]; s += v; s2 += v * v; }
    float m = s * (1.f / D_SZ);
    float var = s2 * (1.f / D_SZ) - m * m;
    st[tid][0] = m; st[tid][1] = rsqrtf(var + EPS_F);
  }
  __syncthreads();
  for (int o = tid; o < P_SZ * D_SZ; o += 256) {
    int p = o >> 8, d = o & 255;
    lng[p][d] = (gft[p][d] - st[p][0]) * st[p][1] * g1[d] + b1[d];
  }
  __syncthreads();
  for (int o = tid; o < P_SZ * INNER; o += 256) {
    int p = o >> 9, i = o & 511;
    float s = 0.f;
    for (int d = 0; d < D_SZ; ++d) s += lng[p][d] * Wq1[d * INNER + i];
    qh[p][i] = s;
  }
  __syncthreads();
  for (int o = tid; o < P_SZ * D_SZ * 16; o += 256) {
    int p = o >> 12, rem = o & 4095, d = rem >> 4, h = rem & 15;
    float s = 0.f;
    if (h < HEADS) {
      for (int c = 0; c < DH; ++c)
        s += Wk1[d * INNER + h * DH + c] * qh[p][h * DH + c];
      s *= SCALE_F;
    }
    U[o] = s;
  }
}

// ---------------------------------------------------------------------------
// K1: one workgroup per (b,p) slice (128 WGs, 8 waves).
//  Phase 0: per-row LN stats (wave reductions, prefetch next rows)
//  Phase 1: dots(2048x8) = LN(X)(2048x256) @ U(256x16)  via WMMA f32 16x16x4
//  Phase 2: softmax over j=2048 per head (in LDS, wave == head)
//  Phase 3: W(16x256)  = A(16x2048) @ LN(X)(2048x256)   via WMMA f32 16x16x4
// Fragment layouts per ISA 7.12.2:
//  A 16x4:  lanes0-15 K={k,k+1}, lanes16-31 K={k+2,k+3}; vgpr = K within pair
//  B 4x16:  mirrored (b.x=B[kk][n], b.y=B[kk+1][n], n=lane&15)
//  C 16x16: vgpr r -> M = r + 8*(lane>=16), N = lane&15
// ---------------------------------------------------------------------------
__global__ __launch_bounds__(256) void k1_attn1(
    const float* __restrict__ x, const float* __restrict__ g1,
    const float* __restrict__ b1, const float* __restrict__ U,
    float* __restrict__ w_out)
{
  extern __shared__ float smem[];
  float* s_stats = smem;                  // 2048*2  (mean, rvar)
  float* s_dots  = s_stats + N_SZ * 2;    // 2048*8
  float* s_U     = s_dots + N_SZ * 8;     // 256*16
  float* s_g     = s_U + D_SZ * 16;       // 256
  float* s_b     = s_g + D_SZ;            // 256

  const int slice = blockIdx.x;           // b*4 + p
  const int p     = slice & 3;
  const float* xs = x + (size_t)slice * N_SZ * D_SZ;
  const int tid   = threadIdx.x;
  const int wave  = tid >> 5;
  const int lane  = tid & 31;
  const int mrow  = lane & 15;
  const int khalf = lane >> 4;

  for (int i = tid; i < D_SZ; i += 256) { s_g[i] = g1[i]; s_b[i] = b1[i]; }
  for (int i = tid; i < D_SZ * 16; i += 256) s_U[i] = U[p * D_SZ * 16 + i];

  // ---- Phase 0: row stats ----
  for (int r = wave; r < N_SZ; r += 8) {
    const float* xr = xs + (size_t)r * D_SZ;
    if (r + 8 < N_SZ)
      __builtin_prefetch(xs + (size_t)(r + 8) * D_SZ + lane * 8, 0, 0);
    float s = 0.f, s2 = 0.f;
    for (int c = lane; c < D_SZ; c += 32) { float v = xr[c]; s += v; s2 += v * v; }
    s = wave_sum(s); s2 = wave_sum(s2);
    if (lane == 0) {
      float m = s * (1.f / D_SZ);
      float var = s2 * (1.f / D_SZ) - m * m;
      s_stats[r * 2 + 0] = m;
      s_stats[r * 2 + 1] = rsqrtf(var + EPS_F);
    }
  }
  __syncthreads();

  // ---- Phase 1: dots = LN(X) @ U ----
  for (int t = wave; t < N_SZ / 16; t += 8) {
    const int r0  = t * 16;
    const int row = r0 + mrow;
    const float mean = s_stats[row * 2 + 0];
    const float rvar = s_stats[row * 2 + 1];
    const float* xr  = xs + (size_t)row * D_SZ;
    v8f acc = {0.f, 0.f, 0.f, 0.f, 0.f, 0.f, 0.f, 0.f};
    for (int k = 0; k < D_SZ; k += 4) {
      const int kk = k + 2 * khalf;
      v2f a, bb;
      a.x = (xr[kk]     - mean) * rvar * s_g[kk]     + s_b[kk];
      a.y = (xr[kk + 1] - mean) * rvar * s_g[kk + 1] + s_b[kk + 1];
      bb.x = s_U[kk * 16 + mrow];
      bb.y = s_U[(kk + 1) * 16 + mrow];
      acc = __builtin_amdgcn_wmma_f32_16x16x4_f32(false, a, false, bb,
                                                  (short)0, acc, false, false);
    }
    if (mrow < HEADS) {
#pragma unroll
      for (int rI = 0; rI < 8; ++rI)
        s_dots[(r0 + rI + 8 * khalf) * 8 + mrow] = acc[rI];
    }
  }
  __syncthreads();

  // ---- Phase 2: softmax over j per head (wave == head) ----
  {
    const int h = wave;
    float m = -3.0e38f;
    for (int j = lane; j < N_SZ; j += 32) m = fmaxf(m, s_dots[j * 8 + h]);
    m = wave_max(m);
    float s = 0.f;
    for (int j = lane; j < N_SZ; j += 32) {
      float e = __expf(s_dots[j * 8 + h] - m);
      s_dots[j * 8 + h] = e;
      s += e;
    }
    s = wave_sum(s);
    const float inv = 1.f / s;
    for (int j = lane; j < N_SZ; j += 32) s_dots[j * 8 + h] *= inv;
  }
  __syncthreads();

  // ---- Phase 3: W = A(16x2048) @ LN(X)(2048x256); wave w -> cols [32w,32w+32) ----
  {
    const int n0 = wave * 32;
    const float hmask = (mrow < HEADS) ? 1.f : 0.f;
    const int hidx = mrow & 7;
    v8f acc0 = {0.f, 0.f, 0.f, 0.f, 0.f, 0.f, 0.f, 0.f};
    v8f acc1 = {0.f, 0.f, 0.f, 0.f, 0.f, 0.f, 0.f, 0.f};
    for (int k = 0; k < N_SZ; k += 4) {
      const int kk = k + 2 * khalf;
      v2f a;
      a.x = s_dots[kk * 8 + hidx] * hmask;
      a.y = s_dots[(kk + 1) * 8 + hidx] * hmask;
      const float m0 = s_stats[kk * 2 + 0], rv0 = s_stats[kk * 2 + 1];
      const float m1 = s_stats[kk * 2 + 2], rv1 = s_stats[kk * 2 + 3];
      {
        const int c = n0 + mrow;
        const float gg = s_g[c], bb = s_b[c];
        v2f bf;
        bf.x = (xs[(size_t)kk * D_SZ + c]       - m0) * rv0 * gg + bb;
        bf.y = (xs[(size_t)(kk + 1) * D_SZ + c] - m1) * rv1 * gg + bb;
        acc0 = __builtin_amdgcn_wmma_f32_16x16x4_f32(false, a, false, bf,
                                                     (short)0, acc0, false, false);
      }
      {
        const int c = n0 + 16 + mrow;
        const float gg = s_g[c], bb = s_b[c];
        v2f bf;
        bf.x = (xs[(size_t)kk * D_SZ + c]       - m0) * rv0 * gg + bb;
        bf.y = (xs[(size_t)(kk + 1) * D_SZ + c] - m1) * rv1 * gg + bb;
        acc1 = __builtin_amdgcn_wmma_f32_16x16x4_f32(false, a, false, bf,
                                                     (short)0, acc1, false, false);
      }
    }
    if (khalf == 0) {  // C rows M=0..7 == the 8 real heads
      float* wo = w_out + (size_t)slice * HEADS * D_SZ;
#pragma unroll
      for (int rI = 0; rI < 8; ++rI) {
        wo[rI * D_SZ + n0 + mrow]      = acc0[rI];
        wo[rI * D_SZ + n0 + 16 + mrow] = acc1[rI];
      }
    }
  }
}

// ---------------------------------------------------------------------------
// K2: the tiny (32,4,256)-scale middle: head V/Wo1, ln2, self-attn over 4
// tokens (softmax over axis=-2 per the reference!), Wo2, ln3, Wv3*Wo3 -> r.
// One block per batch b.
// ---------------------------------------------------------------------------
__global__ __launch_bounds__(256) void k2_mid(
    const float* __restrict__ w_in, const float* __restrict__ gft,
    const float* __restrict__ Wv1, const float* __restrict__ Wo1,
    const float* __restrict__ bo1, const float* __restrict__ g2,
    const float* __restrict__ b2,  const float* __restrict__ Wqkv,
    const float* __restrict__ Wo2, const float* __restrict__ bo2,
    const float* __restrict__ g3,  const float* __restrict__ b3,
    const float* __restrict__ Wv3, const float* __restrict__ Wo3,
    const float* __restrict__ bo3, float* __restrict__ r_out)
{
  __shared__ float ovec[P_SZ][INNER];
  __shared__ float xg[P_SZ][D_SZ];
  __shared__ float lx[P_SZ][D_SZ];
  __shared__ float qkv[P_SZ][3 * INNER];
  __shared__ float att[P_SZ][D_SZ];
  __shared__ float vv[P_SZ][INNER];
  __shared__ float sdots[HEADS][P_SZ][P_SZ];
  __shared__ float st[P_SZ][2];

  const int b = blockIdx.x;
  const int tid = threadIdx.x;

  // o_flat[p][i] = sum_d w[b,p,h(i),d] * Wv1[d][i]
  for (int o = tid; o < P_SZ * INNER; o += 256) {
    int p = o >> 9, i = o & 511, h = i >> 6;
    const float* wr = w_in + (((size_t)(b * P_SZ + p)) * HEADS + h) * D_SZ;
    float s = 0.f;
    for (int d = 0; d < D_SZ; ++d) s += wr[d] * Wv1[d * INNER + i];
    ovec[p][i] = s;
  }
  __syncthreads();
  // xg = o_flat @ Wo1 + bo1 + gft
  for (int o = tid; o < P_SZ * D_SZ; o += 256) {
    int p = o >> 8, e = o & 255;
    float s = bo1[e] + gft[p * D_SZ + e];
    for (int i = 0; i < INNER; ++i) s += ovec[p][i] * Wo1[i * D_SZ + e];
    xg[p][e] = s;
  }
  __syncthreads();
  if (tid < P_SZ) {
    float s = 0.f, s2 = 0.f;
    for (int e = 0; e < D_SZ; ++e) { float v = xg[tid][e]; s += v; s2 += v * v; }
    float m = s * (1.f / D_SZ), var = s2 * (1.f / D_SZ) - m * m;
    st[tid][0] = m; st[tid][1] = rsqrtf(var + EPS_F);
  }
  __syncthreads();
  for (int o = tid; o < P_SZ * D_SZ; o += 256) {
    int p = o >> 8, e = o & 255;
    lx[p][e] = (xg[p][e] - st[p][0]) * st[p][1] * g2[e] + b2[e];
  }
  __syncthreads();
  for (int o = tid; o < P_SZ * 3 * INNER; o += 256) {
    int p = o / (3 * INNER), i = o % (3 * INNER);
    float s = 0.f;
    for (int e = 0; e < D_SZ; ++e) s += lx[p][e] * Wqkv[e * 3 * INNER + i];
    qkv[p][i] = s;
  }
  __syncthreads();
  if (tid < HEADS * P_SZ * P_SZ) {
    int h = tid >> 4, i = (tid >> 2) & 3, j = tid & 3;
    float s = 0.f;
    for (int c = 0; c < DH; ++c)
      s += qkv[i][h * DH + c] * qkv[j][INNER + h * DH + c];
    sdots[h][i][j] = s * SCALE_F;
  }
  __syncthreads();
  if (tid < HEADS * P_SZ) {  // softmax over i (axis=-2) per (h, j)
    int h = tid >> 2, j = tid & 3;
    float m = sdots[h][0][j];
    for (int i = 1; i < P_SZ; ++i) m = fmaxf(m, sdots[h][i][j]);
    float e[P_SZ]; float s = 0.f;
    for (int i = 0; i < P_SZ; ++i) { e[i] = __expf(sdots[h][i][j] - m); s += e[i]; }
    float inv = 1.f / s;
    for (int i = 0; i < P_SZ; ++i) sdots[h][i][j] = e[i] * inv;
  }
  __syncthreads();
  // o2[i][.] = sum_j a[h][i][j] v[j][.]  (reuse ovec)
  for (int o = tid; o < P_SZ * INNER; o += 256) {
    int p = o >> 9, i = o & 511, h = i >> 6;
    float s = 0.f;
    for (int j = 0; j < P_SZ; ++j)
      s += sdots[h][p][j] * qkv[j][2 * INNER + i];
    ovec[p][i] = s;
  }
  __syncthreads();
  for (int o = tid; o < P_SZ * D_SZ; o += 256) {
    int p = o >> 8, e = o & 255;
    float s = bo2[e] + xg[p][e];
    for (int i = 0; i < INNER; ++i) s += ovec[p][i] * Wo2[i * D_SZ + e];
    att[p][e] = s;
  }
  __syncthreads();
  if (tid < P_SZ) {
    float s = 0.f, s2 = 0.f;
    for (int e = 0; e < D_SZ; ++e) { float v = att[tid][e]; s += v; s2 += v * v; }
    float m = s * (1.f / D_SZ), var = s2 * (1.f / D_SZ) - m * m;
    st[tid][0] = m; st[tid][1] = rsqrtf(var + EPS_F);
  }
  __syncthreads();
  for (int o = tid; o < P_SZ * D_SZ; o += 256) {
    int p = o >> 8, e = o & 255;
    lx[p][e] = (att[p][e] - st[p][0]) * st[p][1] * g3[e] + b3[e];
  }
  __syncthreads();
  for (int o = tid; o < P_SZ * INNER; o += 256) {
    int p = o >> 9, i = o & 511;
    float s = 0.f;
    for (int e = 0; e < D_SZ; ++e) s += lx[p][e] * Wv3[e * INNER + i];
    vv[p][i] = s;
  }
  __syncthreads();
  for (int o = tid; o < P_SZ * D_SZ; o += 256) {
    int p = o >> 8, e = o & 255;
    float s = bo3[e];
    for (int i = 0; i < INNER; ++i) s += vv[p][i] * Wo3[i * D_SZ + e];
    r_out[((size_t)b * P_SZ + p) * D_SZ + e] = s;
  }
}

// ---------------------------------------------------------------------------
// K3: out[b,p,n,:] = x[b,p,n,:] + r[b,p,:]  (512MB streamed, float4)
// ---------------------------------------------------------------------------
__global__ __launch_bounds__(256) void k3_add(
    const float* __restrict__ x, const float* __restrict__ r,
    float* __restrict__ out)
{
  const size_t i = (size_t)blockIdx.x * blockDim.x + threadIdx.x;  // float4 idx
  const float4 xv = reinterpret_cast<const float4*>(x)[i];
  const size_t slice = i >> 17;       // 2048*256/4 float4 per (b,p) slice
  const int col4 = (int)(i & 63);     // 256/4 float4 per row
  const float4 rv = reinterpret_cast<const float4*>(r)[slice * 64 + col4];
  float4 o;
  o.x = xv.x + rv.x; o.y = xv.y + rv.y; o.z = xv.z + rv.z; o.w = xv.w + rv.w;
  reinterpret_cast<float4*>(out)[i] = o;
}

// ---------------------------------------------------------------------------
extern "C" void kernel_launch(void* const* d_in, const int* in_sizes, int n_in,
                              void* d_out, int out_size, void* d_ws, size_t ws_size,
                              hipStream_t stream) {
  (void)in_sizes; (void)n_in; (void)out_size; (void)ws_size;
  const float* x    = (const float*)d_in[0];
  const float* gfe  = (const float*)d_in[1];
  const float* Wg   = (const float*)d_in[2];
  const float* bg   = (const float*)d_in[3];
  const float* g1   = (const float*)d_in[4];
  const float* b1   = (const float*)d_in[5];
  const float* Wk1  = (const float*)d_in[6];
  const float* Wv1  = (const float*)d_in[7];
  const float* Wq1  = (const float*)d_in[8];
  const float* Wo1  = (const float*)d_in[9];
  const float* bo1  = (const float*)d_in[10];
  const float* g2   = (const float*)d_in[11];
  const float* b2   = (const float*)d_in[12];
  const float* Wqkv = (const float*)d_in[13];
  const float* Wo2  = (const float*)d_in[14];
  const float* bo2  = (const float*)d_in[15];
  const float* g3   = (const float*)d_in[16];
  const float* b3   = (const float*)d_in[17];
  // d_in[18] (Wk3) and d_in[20] (Wq3) are mathematically dead: the third
  // attention has a single key, so its softmax is identically 1.
  const float* Wv3  = (const float*)d_in[19];
  const float* Wo3  = (const float*)d_in[21];
  const float* bo3  = (const float*)d_in[22];

  float* ws    = (float*)d_ws;
  float* wsU   = ws;                              // 4*256*16      = 16384 f
  float* wsGft = wsU + P_SZ * D_SZ * 16;          // 4*256         =  1024 f
  float* wsW   = wsGft + P_SZ * D_SZ;             // 32*4*8*256    = 262144 f
  float* wsR   = wsW + (size_t)B_SZ * P_SZ * HEADS * D_SZ;  // 32*4*256 = 32768 f
  // total ws: ~1.22 MB

  k0_setup<<<1, 256, 0, stream>>>(gfe, Wg, bg, g1, b1, Wk1, Wq1, wsU, wsGft);

  const size_t smem1 =
      (size_t)(N_SZ * 2 + N_SZ * 8 + D_SZ * 16 + D_SZ + D_SZ) * sizeof(float);
  k1_attn1<<<B_SZ * P_SZ, 256, smem1, stream>>>(x, g1, b1, wsU, wsW);

  k2_mid<<<B_SZ, 256, 0, stream>>>(wsW, wsGft, Wv1, Wo1, bo1, g2, b2, Wqkv,
                                   Wo2, bo2, g3, b3, Wv3, Wo3, bo3, wsR);

  const size_t n4 = (size_t)B_SZ * P_SZ * N_SZ * D_SZ / 4;  // 16,777,216
  k3_add<<<(unsigned)(n4 / 256), 256, 0, stream>>>(x, wsR, (float*)d_out);
}